// GCNConvThr_76690936037710
// MI455X (gfx1250) — compile-verified
//
#include <hip/hip_runtime.h>

typedef __attribute__((ext_vector_type(2))) float v2f;
typedef __attribute__((ext_vector_type(8))) float v8f;

#define THRESH_A 0.4f
#define THRESH_W 0.01f
#define F 64  // F_IN == F_OUT == 64

// ---------------- scratch init ----------------
__global__ void k_zero(float* __restrict__ p, int nwords) {
    int i = blockIdx.x * blockDim.x + threadIdx.x;
    if (i < nwords) p[i] = 0.0f;
}

// ---------------- lock bitmap ----------------
__global__ void k_bitmap(const int* __restrict__ node_lock, int nlock,
                         unsigned* __restrict__ bitmap) {
    int i = blockIdx.x * blockDim.x + threadIdx.x;
    if (i < nlock) {
        int nd = node_lock[i];
        atomicOr(&bitmap[nd >> 5], 1u << (nd & 31));
    }
}

// ---------------- column sum of squares of x ----------------
// block handles 512 rows; thread t owns column t&63, row phase t>>6 (stride 4)
__global__ void k_colnorm(const float* __restrict__ x, int n, float* __restrict__ coln) {
    int col  = threadIdx.x & 63;
    int base = blockIdx.x * 512;
    int rend = base + 512; if (rend > n) rend = n;
    float s = 0.0f;
    for (int r = base + (threadIdx.x >> 6); r < rend; r += 4) {
        float v = x[(size_t)r * F + col];
        s += v * v;
    }
    atomicAdd(&coln[col], s);
}

// ---------------- weight pruning: |W| < 0.01/||x_col||  <=>  |W|*||x_col|| < 0.01
__global__ void k_prunew(const float* __restrict__ W, const float* __restrict__ coln,
                         float* __restrict__ Wp) {
    int i = blockIdx.x * blockDim.x + threadIdx.x;
    if (i < F * F) {
        int kcol = i & 63;
        float w  = W[i];
        Wp[i] = (fabsf(w) * sqrtf(coln[kcol]) < THRESH_W) ? 0.0f : w;
    }
}

// ---------------- h = x @ Wp^T via V_WMMA_F32_16X16X4_F32 ----------------
// one wave per 16x16 output tile; K=64 in 16 steps of 4
__global__ void k_gemm(const float* __restrict__ x, const float* __restrict__ Wp,
                       float* __restrict__ h, int n) {
    int wave = (blockIdx.x * blockDim.x + threadIdx.x) >> 5;
    int ntiles = (n >> 4) * (F / 16);
    if (wave >= ntiles) return;
    int lid  = threadIdx.x & 31;
    int rt   = wave >> 2;          // row tile (16 nodes)
    int ct   = wave & 3;           // col tile (16 outputs)
    int row0 = rt * 16, col0 = ct * 16;
    int half = lid >> 4;           // 0: K pair {0,1}, 1: K pair {2,3}
    int l15  = lid & 15;

    const float* arow = x  + (size_t)(row0 + l15) * F;   // A: M = l15
    const float* brow = Wp + (size_t)(col0 + l15) * F;   // B: N = l15 (B = Wp^T)

    v8f c = {0.f, 0.f, 0.f, 0.f, 0.f, 0.f, 0.f, 0.f};
#pragma unroll
    for (int k = 0; k < F; k += 4) {
        v2f a = *(const v2f*)(arow + k + 2 * half);   // K = k+2*half, k+2*half+1
        v2f b = *(const v2f*)(brow + k + 2 * half);
        c = __builtin_amdgcn_wmma_f32_16x16x4_f32(
                /*neg_a=*/false, a, /*neg_b=*/false, b,
                /*c_mod=*/(short)0, c, /*reuse_a=*/false, /*reuse_b=*/false);
    }
    // C/D layout: VGPR r, lanes 0-15 -> M=r, lanes 16-31 -> M=r+8; N = l15
    float* hp = h + (size_t)row0 * F + col0 + l15;
#pragma unroll
    for (int r = 0; r < 8; ++r)
        hp[(size_t)(r + 8 * half) * F] = c[r];
}

// ---------------- out = bias (broadcast) ----------------
__global__ void k_initout(float* __restrict__ out, const float* __restrict__ bias, long long total) {
    long long i = (long long)blockIdx.x * blockDim.x + threadIdx.x;
    if (i < total) out[i] = bias[i & 63];
}

// ---------------- per-edge feature norms + global sum of squares ----------------
// one wave per edge; lane handles features {2l, 2l+1}
__global__ void k_edgenorm(const float* __restrict__ h, const int* __restrict__ src,
                           const float* __restrict__ ew, float* __restrict__ nf,
                           float* __restrict__ norm2, long long m) {
    __shared__ float s_sum;
    if (threadIdx.x == 0) s_sum = 0.0f;
    __syncthreads();

    long long e = (long long)blockIdx.x * 8 + (threadIdx.x >> 5);
    int lid = threadIdx.x & 31;
    if (e < m) {
        int sn = src[e];
        v2f v = *(const v2f*)(h + (size_t)sn * F + 2 * lid);
        float p = v.x * v.x + v.y * v.y;
#pragma unroll
        for (int o = 16; o; o >>= 1) p += __shfl_xor(p, o, 32);
        if (lid == 0) {
            float val = sqrtf(p) * fabsf(ew[e]);   // ||w * h_src|| = |w| * ||h_src||
            nf[e] = val;
            atomicAdd(&s_sum, val * val);          // LDS ds_add_f32
        }
    }
    __syncthreads();
    if (threadIdx.x == 0 && s_sum != 0.0f) unsafeAtomicAdd(norm2, s_sum);
}

// ---------------- masked scatter-add + keep mask ----------------
__global__ void k_edgescatter(const float* __restrict__ h, const int* __restrict__ src,
                              const int* __restrict__ dst, const float* __restrict__ ew,
                              const float* __restrict__ nf, const float* __restrict__ norm2,
                              const unsigned* __restrict__ bitmap,
                              float* __restrict__ out, float* __restrict__ keep, long long m) {
    long long e = (long long)blockIdx.x * 8 + (threadIdx.x >> 5);
    if (e >= m) return;
    int lid = threadIdx.x & 31;
    int s = src[e], d = dst[e];

    bool lock = (((bitmap[d >> 5] >> (d & 31)) & 1u) != 0u) || (s == d);
    float na = sqrtf(*norm2);
    bool pruned = ((nf[e] * (1.0f / (float)F)) < (THRESH_A * na / (float)m)) && !lock;

    if (lid == 0) keep[e] = pruned ? 0.0f : 1.0f;
    if (!pruned) {
        float w = ew[e];
        v2f v = *(const v2f*)(h + (size_t)s * F + 2 * lid);
        float* o = out + (size_t)d * F + 2 * lid;
        unsafeAtomicAdd(o + 0, w * v.x);   // global_atomic_add_f32, hits L2 atomic units
        unsafeAtomicAdd(o + 1, w * v.y);
    }
}

extern "C" void kernel_launch(void* const* d_in, const int* in_sizes, int n_in,
                              void* d_out, int out_size, void* d_ws, size_t ws_size,
                              hipStream_t stream) {
    const float* x         = (const float*)d_in[0];
    const int*   edge_idx  = (const int*)  d_in[1];
    const float* ew        = (const float*)d_in[2];
    const int*   node_lock = (const int*)  d_in[3];
    const float* W         = (const float*)d_in[4];
    const float* bias      = (const float*)d_in[5];

    const int       n = in_sizes[0] / F;        // 100000 nodes
    const long long m = (long long)in_sizes[2]; // 1.6M edges
    const int   nlock = in_sizes[3];

    const int* src = edge_idx;              // edge_index[0]
    const int* dst = edge_idx + m;          // edge_index[1]

    // workspace layout (float offsets)
    float*    ws_f   = (float*)d_ws;
    float*    norm2  = ws_f;                       // [0]
    float*    coln   = ws_f + 64;                  // [64..127]
    unsigned* bitmap = (unsigned*)(ws_f + 128);    // (n+31)/32 words
    float*    Wp     = ws_f + 4096;                // 64x64
    float*    h      = ws_f + 8192;                // n*64
    float*    nf     = h + (size_t)n * F;          // m

    float* out  = (float*)d_out;                   // n*64
    float* keep = out + (size_t)n * F;             // m

    const int bitwords  = (n + 31) / 32;
    const int zerowords = 128 + bitwords;

    k_zero<<<(zerowords + 255) / 256, 256, 0, stream>>>(ws_f, zerowords);
    k_bitmap<<<(nlock + 255) / 256, 256, 0, stream>>>(node_lock, nlock, bitmap);
    k_colnorm<<<(n + 511) / 512, 256, 0, stream>>>(x, n, coln);
    k_prunew<<<(F * F + 255) / 256, 256, 0, stream>>>(W, coln, Wp);

    const int ntiles = (n / 16) * (F / 16);        // one wave per 16x16 tile
    k_gemm<<<(ntiles + 7) / 8, 256, 0, stream>>>(x, Wp, h, n);

    const long long total = (long long)n * F;
    k_initout<<<(int)((total + 255) / 256), 256, 0, stream>>>(out, bias, total);

    const int eblocks = (int)((m + 7) / 8);        // 8 waves (edges) per block
    k_edgenorm<<<eblocks, 256, 0, stream>>>(h, src, ew, nf, norm2, m);
    k_edgescatter<<<eblocks, 256, 0, stream>>>(h, src, dst, ew, nf, norm2, bitmap,
                                               out, keep, m);
}